// HashTableVoxelizedGaussianAdapterModule_30047591203014
// MI455X (gfx1250) — compile-verified
//
#include <hip/hip_runtime.h>
#include <hip/hip_bf16.h>
#include <stdint.h>

#define TBL_T 524288u              // hash table entries (power of two)
#define NVOX  (128u * 128u * 128u) // 2,097,152 voxels

typedef float v2f __attribute__((ext_vector_type(2)));
typedef float v4f __attribute__((ext_vector_type(4)));
typedef float v8f __attribute__((ext_vector_type(8)));

// sigmoid(v - 4) > mo  <=>  v > 4 + ln(mo/(1-mo))
#define TH0 0.10817970f   // mo = 0.02
#define TH1 1.05556102f   // mo = 0.05
#define TH2 1.80277542f   // mo = 0.10

__device__ __forceinline__ uint32_t hash3(uint32_t x, uint32_t y, uint32_t z) {
    return (x ^ (y * 2654435761u) ^ (z * 805459861u)) & (TBL_T - 1u);
}

// Full-wave (32 lane) sum using V_WMMA_F32_16X16X4_F32 with B == ones.
// A layout (16x4 f32): lanes 0-15 hold A[m][0],A[m][1]; lanes 16-31 hold A[m][2],A[m][3].
// With a = {v, 0}: row-sum(m) = v_m + v_{m+16}, replicated across the D columns.
// D VGPR i holds rowsum(i) in lanes 0-15 and rowsum(i+8) in lanes 16-31, so
// sum(d[0..7]) gives each half-wave's 16-lane sum; shfl_xor(16) completes it.
// Must be called with EXEC all-ones (fully converged, full wave).
__device__ __forceinline__ float wave_sum_wmma(float v) {
    v2f a; a[0] = v;    a[1] = 0.0f;
    v2f b; b[0] = 1.0f; b[1] = 1.0f;
    v8f c = {};
    v8f d = __builtin_amdgcn_wmma_f32_16x16x4_f32(false, a, false, b,
                                                  (short)0, c, false, false);
    float t = ((d[0] + d[1]) + (d[2] + d[3])) + ((d[4] + d[5]) + (d[6] + d[7]));
    t += __shfl_xor(t, 16, 32);
    return t;
}

__global__ void k_init(float* __restrict__ ws) {
    if (threadIdx.x < 4) ws[threadIdx.x] = 0.0f;
}

// Pass 1: masked sum / sumsq / count over feature channels 0..2
__global__ void k_stats(const float* __restrict__ t0,
                        const float* __restrict__ t1,
                        const float* __restrict__ t2,
                        float* __restrict__ ws) {
    const float* __restrict__ t0o = t0 + (size_t)13 * TBL_T;
    const float* __restrict__ t1o = t1 + (size_t)13 * TBL_T;
    float s = 0.0f, ss = 0.0f, cnt = 0.0f;

    const uint32_t stride = gridDim.x * blockDim.x;
    for (uint32_t n = blockIdx.x * blockDim.x + threadIdx.x; n < NVOX; n += stride) {
        const uint32_t x = n >> 14, y = (n >> 7) & 127u, z = n & 127u;
        bool m = t0o[hash3(x >> 4, y >> 4, z >> 4)] > TH0;
        if (m) m = t1o[hash3(x >> 2, y >> 2, z >> 2)] > TH1;
        if (m) {
            const uint32_t idx = hash3(x, y, z);
            if (t2[(size_t)13 * TBL_T + idx] > TH2) {
                const float a = t2[idx];
                const float b = t2[(size_t)TBL_T + idx];
                const float d = t2[(size_t)2 * TBL_T + idx];
                s   += a + b + d;
                ss  += a * a + b * b + d * d;
                cnt += 1.0f;
            }
        }
    }
    // fully converged here: WMMA wave reduction
    s   = wave_sum_wmma(s);
    ss  = wave_sum_wmma(ss);
    cnt = wave_sum_wmma(cnt);

    __shared__ float ls[8][3];
    const int wid  = threadIdx.x >> 5;
    const int lane = threadIdx.x & 31;
    if (lane == 0) { ls[wid][0] = s; ls[wid][1] = ss; ls[wid][2] = cnt; }
    __syncthreads();
    if (threadIdx.x == 0) {
        float S = 0.0f, SS = 0.0f, C = 0.0f;
        #pragma unroll
        for (int w = 0; w < 8; ++w) { S += ls[w][0]; SS += ls[w][1]; C += ls[w][2]; }
        atomicAdd(&ws[0], S);
        atomicAdd(&ws[1], SS);
        atomicAdd(&ws[2], C);
    }
}

// Pass 2: fold raw sums into mean / inv_std; precompute camera mean + far scales
__global__ void k_finalize(const float* __restrict__ extr,
                           const float* __restrict__ fars,
                           float* __restrict__ ws) {
    if (threadIdx.x == 0 && blockIdx.x == 0) {
        const float S = ws[0], SS = ws[1], Cv = ws[2];
        const float cnt  = Cv * 3.0f;
        const float mean = S / cnt;
        const float var  = (SS - cnt * mean * mean) / fmaxf(cnt - 1.0f, 1.0f);
        ws[4] = mean;
        ws[5] = rsqrtf(var);
        #pragma unroll
        for (int k = 0; k < 3; ++k)
            ws[6 + k] = 0.25f * (extr[0 * 16 + k * 4 + 3] + extr[1 * 16 + k * 4 + 3] +
                                 extr[2 * 16 + k * 4 + 3] + extr[3 * 16 + k * 4 + 3]);
        const float far = fars[0];
        ws[9]  = far;
        ws[10] = 2.0f * far / 128.0f;          // voxel-center scale
        ws[11] = 2.0f * far / 128.0f / 3.0f;   // normalized-mean scale
    }
}

__device__ __forceinline__ void nt_store(float* p, float v) {
    __builtin_nontemporal_store(v, p);
}
__device__ __forceinline__ void nt_store4z(v4f* p) {
    v4f z = {0.0f, 0.0f, 0.0f, 0.0f};
    __builtin_nontemporal_store(z, p);
}

// Pass 3: per-voxel outputs (means | cov | harm | opac), zeros on masked voxels.
// All output stores are non-temporal so the 128 MiB output stream does not
// evict the (L2-resident, 84 MB) hash tables from the 192 MB L2.
__global__ void __launch_bounds__(256)
k_main(const float* __restrict__ t0,
       const float* __restrict__ t1,
       const float* __restrict__ t2,
       const float* __restrict__ ws,
       float* __restrict__ out) {
    const uint32_t n    = blockIdx.x * blockDim.x + threadIdx.x;
    const uint32_t lane = threadIdx.x & 31u;
    const uint32_t x = n >> 14, y = (n >> 7) & 127u, z = n & 127u;

    float* __restrict__ means = out;
    float* __restrict__ cov   = out + (size_t)3  * NVOX;
    float* __restrict__ harm  = out + (size_t)12 * NVOX;
    float* __restrict__ opac  = out + (size_t)15 * NVOX;

    const float* __restrict__ t0o = t0 + (size_t)13 * TBL_T;
    const float* __restrict__ t1o = t1 + (size_t)13 * TBL_T;

    bool m = t0o[hash3(x >> 4, y >> 4, z >> 4)] > TH0;
    if (m) m = t1o[hash3(x >> 2, y >> 2, z >> 2)] > TH1;
    const uint32_t idx = hash3(x, y, z);
    float f13 = 0.0f;
    if (m) {
        f13 = t2[(size_t)13 * TBL_T + idx];
        m = f13 > TH2;
    }

    // ---- Fast path: whole wave masked out (~93% of waves) ----
    // Cooperative wide zero fill: per wave, means=24, cov=72, harm=24, opac=8
    // 16-byte chunks, all bases 16B aligned (wave base voxel is a multiple of 32).
    if (__ballot(m) == 0ull) {
        const uint32_t n0 = n - lane;              // wave's base voxel
        v4f* pm = (v4f*)(means + (size_t)3 * n0);
        v4f* pc = (v4f*)(cov   + (size_t)9 * n0);
        v4f* ph = (v4f*)(harm  + (size_t)3 * n0);
        v4f* po = (v4f*)(opac  + (size_t)n0);
        if (lane < 24) nt_store4z(pm + lane);
        nt_store4z(pc + lane);
        nt_store4z(pc + lane + 32);
        if (lane < 8) nt_store4z(pc + lane + 64);
        if (lane < 24) nt_store4z(ph + lane);
        if (lane < 8)  nt_store4z(po + lane);
        return;
    }

    // ---- Mixed wave: per-thread path ----
    if (!m) {
        #pragma unroll
        for (int k = 0; k < 3; ++k) nt_store(&means[(size_t)3 * n + k], 0.0f);
        #pragma unroll
        for (int k = 0; k < 9; ++k) nt_store(&cov[(size_t)9 * n + k], 0.0f);
        #pragma unroll
        for (int k = 0; k < 3; ++k) nt_store(&harm[(size_t)3 * n + k], 0.0f);
        nt_store(&opac[n], 0.0f);
        return;
    }

    float f[13];
    #pragma unroll
    for (int c = 0; c < 13; ++c) f[c] = t2[(size_t)c * TBL_T + idx];

    const float mean = ws[4], inv_std = ws[5];
    const float cam0 = ws[6], cam1 = ws[7], cam2 = ws[8];
    const float far = ws[9], vcs = ws[10], sf = ws[11];

    // means = (dm - mean)/std * (2*far/128/3) + voxel_center
    const float vcx = (float)x * vcs - far + cam0;
    const float vcy = (float)y * vcs - far + cam1;
    const float vcz = (float)z * vcs - far + cam2;
    nt_store(&means[(size_t)3 * n + 0], (f[0] - mean) * inv_std * sf + vcx);
    nt_store(&means[(size_t)3 * n + 1], (f[1] - mean) * inv_std * sf + vcy);
    nt_store(&means[(size_t)3 * n + 2], (f[2] - mean) * inv_std * sf + vcz);

    // quaternion -> rotation, L = R * scale, cov = L L^T
    float qr = f[3], qx = f[4], qy = f[5], qz = f[6];
    const float qinv = rsqrtf(qr * qr + qx * qx + qy * qy + qz * qz);
    qr *= qinv; qx *= qinv; qy *= qinv; qz *= qinv;
    const float sx = f[7], sy = f[8], sz = f[9];

    const float R00 = 1.0f - 2.0f * (qy * qy + qz * qz);
    const float R01 = 2.0f * (qx * qy - qr * qz);
    const float R02 = 2.0f * (qx * qz + qr * qy);
    const float R10 = 2.0f * (qx * qy + qr * qz);
    const float R11 = 1.0f - 2.0f * (qx * qx + qz * qz);
    const float R12 = 2.0f * (qy * qz - qr * qx);
    const float R20 = 2.0f * (qx * qz - qr * qy);
    const float R21 = 2.0f * (qy * qz + qr * qx);
    const float R22 = 1.0f - 2.0f * (qx * qx + qy * qy);

    const float L00 = R00 * sx, L01 = R01 * sy, L02 = R02 * sz;
    const float L10 = R10 * sx, L11 = R11 * sy, L12 = R12 * sz;
    const float L20 = R20 * sx, L21 = R21 * sy, L22 = R22 * sz;

    const float c00 = L00 * L00 + L01 * L01 + L02 * L02;
    const float c01 = L00 * L10 + L01 * L11 + L02 * L12;
    const float c02 = L00 * L20 + L01 * L21 + L02 * L22;
    const float c11 = L10 * L10 + L11 * L11 + L12 * L12;
    const float c12 = L10 * L20 + L11 * L21 + L12 * L22;
    const float c22 = L20 * L20 + L21 * L21 + L22 * L22;

    float* __restrict__ cv = cov + (size_t)9 * n;
    nt_store(&cv[0], c00); nt_store(&cv[1], c01); nt_store(&cv[2], c02);
    nt_store(&cv[3], c01); nt_store(&cv[4], c11); nt_store(&cv[5], c12);
    nt_store(&cv[6], c02); nt_store(&cv[7], c12); nt_store(&cv[8], c22);

    nt_store(&harm[(size_t)3 * n + 0], f[10]);
    nt_store(&harm[(size_t)3 * n + 1], f[11]);
    nt_store(&harm[(size_t)3 * n + 2], f[12]);

    nt_store(&opac[n], 1.0f / (1.0f + __expf(4.0f - f13)));
}

extern "C" void kernel_launch(void* const* d_in, const int* in_sizes, int n_in,
                              void* d_out, int out_size, void* d_ws, size_t ws_size,
                              hipStream_t stream) {
    const float* t0   = (const float*)d_in[0]; // table0 (1,14,T)
    const float* t1   = (const float*)d_in[1]; // table1 (1,14,T)
    const float* t2   = (const float*)d_in[2]; // table2 (1,14,T)
    const float* extr = (const float*)d_in[3]; // extrinsics (1,4,4,4)
    const float* fars = (const float*)d_in[4]; // fars (1,4)
    float* out = (float*)d_out;
    float* ws  = (float*)d_ws;

    k_init<<<1, 32, 0, stream>>>(ws);
    k_stats<<<1024, 256, 0, stream>>>(t0, t1, t2, ws);
    k_finalize<<<1, 32, 0, stream>>>(extr, fars, ws);
    k_main<<<NVOX / 256, 256, 0, stream>>>(t0, t1, t2, ws, out);
}